// SAM2_11123965296901
// MI455X (gfx1250) — compile-verified
//
#include <hip/hip_runtime.h>
#include <hip/hip_bf16.h>

typedef __attribute__((ext_vector_type(2))) float v2f;
typedef __attribute__((ext_vector_type(8))) float v8f;

#define NPIX 3136
#define WID  56
#define EPSF 1e-5f

static __device__ __forceinline__ v8f wmma4(v2f a, v2f b, v8f c) {
  // D = A(16x4 f32) * B(4x16 f32) + C(16x16 f32)
  return __builtin_amdgcn_wmma_f32_16x16x4_f32(false, a, false, b, (short)0, c,
                                               false, false);
}

// ---------------- small elementwise helpers ----------------
__global__ void prep_bn1(const float* __restrict__ g1, const float* __restrict__ bt1,
                         const float* __restrict__ m1, const float* __restrict__ v1,
                         float* __restrict__ s1, float* __restrict__ t1) {
  int k = blockIdx.x * blockDim.x + threadIdx.x;
  if (k >= 3200) return;
  float s = g1[k] * rsqrtf(v1[k] + EPSF);
  s1[k] = s;
  t1[k] = bt1[k] - m1[k] * s;
}

__global__ void zero_f32(float* __restrict__ p, int n) {
  int i = blockIdx.x * blockDim.x + threadIdx.x;
  if (i < n) p[i] = 0.0f;
}

__global__ void bn2_relu(const float* __restrict__ h1, const float* __restrict__ g2,
                         const float* __restrict__ bt2, const float* __restrict__ m2,
                         const float* __restrict__ v2p, float* __restrict__ h2) {
  int i = blockIdx.x * blockDim.x + threadIdx.x;
  if (i >= 2 * 32 * NPIX) return;
  int j = (i / NPIX) & 31;
  float s = g2[j] * rsqrtf(v2p[j] + EPSF);
  h2[i] = fmaxf((h1[i] - m2[j]) * s + bt2[j], 0.0f);
}

// --------- 1x1-conv GEMM, 32(M)x(16*NT)(N) per wave: Y = W[O,C] @ X + bias -----
template <int NT>
__global__ void conv1x1_wmma(const float* __restrict__ X, const float* __restrict__ W,
                             const float* __restrict__ bias, float* __restrict__ Y,
                             int C, int O) {
  int gwave = (blockIdx.x * blockDim.x + threadIdx.x) >> 5;
  int lane  = threadIdx.x & 31;
  const int NTP = NPIX / (16 * NT);      // n-tiles per batch
  const int ntiles = 2 * NTP;
  int mtiles = O / 32;
  if (gwave >= mtiles * ntiles) return;  // wave-uniform exit
  int mt = gwave / ntiles, nt = gwave % ntiles;
  int b = nt / NTP;
  int n0 = (nt % NTP) * (16 * NT);
  const float* Xb = X + (size_t)b * C * NPIX;
  float* Yb = Y + (size_t)b * O * NPIX;
  int half = lane >> 4, l16 = lane & 15;
  int m0 = mt * 32;
  v8f acc0[NT], acc1[NT];
#pragma unroll
  for (int j = 0; j < NT; ++j) { v8f z = {}; acc0[j] = z; acc1[j] = z; }
  for (int k0 = 0; k0 < C; k0 += 4) {
    int ka = k0 + 2 * half;              // K slots ka, ka+1 for this lane
    v2f a0, a1;
    a0.x = W[(size_t)(m0 + l16) * C + ka];
    a0.y = W[(size_t)(m0 + l16) * C + ka + 1];
    a1.x = W[(size_t)(m0 + 16 + l16) * C + ka];
    a1.y = W[(size_t)(m0 + 16 + l16) * C + ka + 1];
#pragma unroll
    for (int j = 0; j < NT; ++j) {
      v2f bf;
      bf.x = Xb[(size_t)ka * NPIX + n0 + 16 * j + l16];
      bf.y = Xb[(size_t)(ka + 1) * NPIX + n0 + 16 * j + l16];
      acc0[j] = wmma4(a0, bf, acc0[j]);
      acc1[j] = wmma4(a1, bf, acc1[j]);
    }
  }
#pragma unroll
  for (int j = 0; j < NT; ++j)
#pragma unroll
    for (int r = 0; r < 8; ++r) {
      int ma = m0 + r + 8 * half;
      int mb = m0 + 16 + r + 8 * half;
      Yb[(size_t)ma * NPIX + n0 + 16 * j + l16] = acc0[j][r] + bias[ma];
      Yb[(size_t)mb * NPIX + n0 + 16 * j + l16] = acc1[j][r] + bias[mb];
    }
}

// ------- stage 2: h1[b,32,N] += cw1[32,3200] @ relu(bn1(cat)) , split-K --------
// cat row k: k<64 -> x1[k];  k>=64 -> x2[c] shifted by p, k = 64 + c*49 + p
// 32(M) x 32(N) per wave, 10-way split-K over the 50 position-blocks.
// All gathers are unconditional from a clamped address; zero-padding applied
// via value-select (v_cndmask), so the hot loop stays branch-free.
__global__ void h1_gemm(const float* __restrict__ x1, const float* __restrict__ x2,
                        const float* __restrict__ cw1, const float* __restrict__ s1,
                        const float* __restrict__ t1, float* __restrict__ h1) {
  int gwave = (blockIdx.x * blockDim.x + threadIdx.x) >> 5;
  int lane  = threadIdx.x & 31;
  const int NSTRIP = 2 * NPIX / 32;      // 196 strips of 32 pixels
  const int NCHUNK = 10;
  if (gwave >= NSTRIP * NCHUNK) return;  // wave-uniform exit
  int strip = gwave / NCHUNK, chunk = gwave % NCHUNK;
  int b = strip / (NPIX / 32);
  int n0 = (strip % (NPIX / 32)) * 32;
  int half = lane >> 4, l16 = lane & 15;
  int nA = n0 + l16, nB = n0 + 16 + l16;
  int yA = nA / WID, xA = nA % WID;
  int yB = nB / WID, xB = nB % WID;
  const float* x1b = x1 + (size_t)b * 64 * NPIX;
  const float* x2b = x2 + (size_t)b * 64 * NPIX;
  v8f a00 = {}, a01 = {}, a10 = {}, a11 = {};  // [m-tile][n-tile]
  for (int pp = chunk * 5; pp < chunk * 5 + 5; ++pp) {
    if (pp == 0) {
      // ---- x1 block: cat rows 0..63, no shift, always in-bounds ----
      for (int c0 = 0; c0 < 64; c0 += 4) {
        int ca = c0 + 2 * half;
        float sc0 = s1[ca], sh0 = t1[ca];
        float sc1 = s1[ca + 1], sh1 = t1[ca + 1];
        v2f bfA, bfB, a0, a1;
        bfA.x = fmaxf(x1b[(size_t)ca * NPIX + nA] * sc0 + sh0, 0.0f);
        bfA.y = fmaxf(x1b[(size_t)(ca + 1) * NPIX + nA] * sc1 + sh1, 0.0f);
        bfB.x = fmaxf(x1b[(size_t)ca * NPIX + nB] * sc0 + sh0, 0.0f);
        bfB.y = fmaxf(x1b[(size_t)(ca + 1) * NPIX + nB] * sc1 + sh1, 0.0f);
        a0.x = cw1[(size_t)l16 * 3200 + ca];
        a0.y = cw1[(size_t)l16 * 3200 + ca + 1];
        a1.x = cw1[(size_t)(16 + l16) * 3200 + ca];
        a1.y = cw1[(size_t)(16 + l16) * 3200 + ca + 1];
        a00 = wmma4(a0, bfA, a00);
        a01 = wmma4(a0, bfB, a01);
        a10 = wmma4(a1, bfA, a10);
        a11 = wmma4(a1, bfB, a11);
      }
    } else {
      // ---- x2 block at position p: cat row k = 64 + c*49 + p ----
      int p = pp - 1;
      int dy = p / 7 - 3, dx = p % 7 - 3;
      int yyA = yA + dy, xxA = xA + dx;
      int yyB = yB + dy, xxB = xB + dx;
      bool inbA = (yyA >= 0) & (yyA < WID) & (xxA >= 0) & (xxA < WID);
      bool inbB = (yyB >= 0) & (yyB < WID) & (xxB >= 0) & (xxB < WID);
      int nnA = inbA ? yyA * WID + xxA : nA;   // clamped, always valid
      int nnB = inbB ? yyB * WID + xxB : nB;
      for (int c0 = 0; c0 < 64; c0 += 4) {
        int ca = c0 + 2 * half;
        int k0 = 64 + ca * 49 + p;
        int k1 = k0 + 49;
        float sc0 = s1[k0], sh0 = t1[k0];
        float sc1 = s1[k1], sh1 = t1[k1];
        float rA0 = x2b[(size_t)ca * NPIX + nnA];        // unconditional loads
        float rA1 = x2b[(size_t)(ca + 1) * NPIX + nnA];
        float rB0 = x2b[(size_t)ca * NPIX + nnB];
        float rB1 = x2b[(size_t)(ca + 1) * NPIX + nnB];
        rA0 = inbA ? rA0 : 0.0f;                          // branchless pad
        rA1 = inbA ? rA1 : 0.0f;
        rB0 = inbB ? rB0 : 0.0f;
        rB1 = inbB ? rB1 : 0.0f;
        v2f bfA, bfB, a0, a1;
        bfA.x = fmaxf(rA0 * sc0 + sh0, 0.0f);
        bfA.y = fmaxf(rA1 * sc1 + sh1, 0.0f);
        bfB.x = fmaxf(rB0 * sc0 + sh0, 0.0f);
        bfB.y = fmaxf(rB1 * sc1 + sh1, 0.0f);
        a0.x = cw1[(size_t)l16 * 3200 + k0];
        a0.y = cw1[(size_t)l16 * 3200 + k1];
        a1.x = cw1[(size_t)(16 + l16) * 3200 + k0];
        a1.y = cw1[(size_t)(16 + l16) * 3200 + k1];
        a00 = wmma4(a0, bfA, a00);
        a01 = wmma4(a0, bfB, a01);
        a10 = wmma4(a1, bfA, a10);
        a11 = wmma4(a1, bfB, a11);
      }
    }
  }
  float* h1b = h1 + (size_t)b * 32 * NPIX;
#pragma unroll
  for (int r = 0; r < 8; ++r) {
    int ma = r + 8 * half, mb = 16 + r + 8 * half;
    atomicAdd(&h1b[(size_t)ma * NPIX + nA], a00[r]);
    atomicAdd(&h1b[(size_t)ma * NPIX + nB], a01[r]);
    atomicAdd(&h1b[(size_t)mb * NPIX + nA], a10[r]);
    atomicAdd(&h1b[(size_t)mb * NPIX + nB], a11[r]);
  }
}

// ------- stage 4: out[b,o,n] = sum_k x3[b,o,n+off(k)] * wgt[b,(o/8)*49+k,n] ----
__global__ void aggregate(const float* __restrict__ x3, const float* __restrict__ wgt,
                          float* __restrict__ out) {
  int i = blockIdx.x * blockDim.x + threadIdx.x;
  if (i >= 2 * 256 * NPIX) return;
  int n = i % NPIX;
  int o = (i / NPIX) % 256;
  int b = i / (256 * NPIX);
  int y = n / WID, x = n % WID, g = o >> 3;
  const float* x3c = x3 + ((size_t)b * 256 + o) * NPIX;
  const float* wg  = wgt + ((size_t)b * 1568 + (size_t)g * 49) * NPIX + n;
  float acc = 0.0f;
#pragma unroll
  for (int k = 0; k < 49; ++k) {
    int yy = y + k / 7 - 3, xx = x + k % 7 - 3;
    bool inb = (yy >= 0) & (yy < WID) & (xx >= 0) & (xx < WID);
    int idx = inb ? yy * WID + xx : n;              // clamped, always valid
    float v = x3c[idx];                             // unconditional load
    v = inb ? v : 0.0f;                             // branchless zero-pad
    acc += v * wg[(size_t)k * NPIX];
  }
  out[i] = acc;
}

// -------------------------------- launch ---------------------------------------
extern "C" void kernel_launch(void* const* d_in, const int* in_sizes, int n_in,
                              void* d_out, int out_size, void* d_ws, size_t ws_size,
                              hipStream_t stream) {
  const float* feat1 = (const float*)d_in[0];
  const float* feat2 = (const float*)d_in[1];
  const float* w1 = (const float*)d_in[2];
  const float* b1 = (const float*)d_in[3];
  const float* w2 = (const float*)d_in[4];
  const float* b2 = (const float*)d_in[5];
  const float* w3 = (const float*)d_in[6];
  const float* b3 = (const float*)d_in[7];
  const float* g1 = (const float*)d_in[8];
  const float* bt1 = (const float*)d_in[9];
  const float* m1 = (const float*)d_in[10];
  const float* v1 = (const float*)d_in[11];
  const float* cw1 = (const float*)d_in[12];
  const float* g2 = (const float*)d_in[13];
  const float* bt2 = (const float*)d_in[14];
  const float* m2 = (const float*)d_in[15];
  const float* v2p = (const float*)d_in[16];
  const float* cw2 = (const float*)d_in[17];
  const float* cb2 = (const float*)d_in[18];
  float* out = (float*)d_out;

  float* ws = (float*)d_ws;
  float* x1  = ws;                          // 2*64*3136
  float* x2  = x1 + (size_t)2 * 64 * NPIX;
  float* x3  = x2 + (size_t)2 * 64 * NPIX;  // 2*256*3136
  float* h1  = x3 + (size_t)2 * 256 * NPIX; // 2*32*3136
  float* h2  = h1 + (size_t)2 * 32 * NPIX;
  float* wgt = h2 + (size_t)2 * 32 * NPIX;  // 2*1568*3136
  float* s1  = wgt + (size_t)2 * 1568 * NPIX;
  float* t1  = s1 + 3200;

  // fused BN1 scale/shift
  prep_bn1<<<(3200 + 255) / 256, 256, 0, stream>>>(g1, bt1, m1, v1, s1, t1);
  // x1, x2, x3 via WMMA 1x1-conv GEMMs (32x64 per wave)
  { int waves = (64 / 32) * (2 * NPIX / 64);
    conv1x1_wmma<4><<<(waves * 32 + 255) / 256, 256, 0, stream>>>(feat1, w1, b1, x1, 256, 64);
    conv1x1_wmma<4><<<(waves * 32 + 255) / 256, 256, 0, stream>>>(feat2, w2, b2, x2, 256, 64); }
  { int waves = (256 / 32) * (2 * NPIX / 64);
    conv1x1_wmma<4><<<(waves * 32 + 255) / 256, 256, 0, stream>>>(feat1, w3, b3, x3, 256, 256); }
  // h1 = cw1 @ relu(bn1(cat))  (split-K, atomic accumulate)
  zero_f32<<<(2 * 32 * NPIX + 255) / 256, 256, 0, stream>>>(h1, 2 * 32 * NPIX);
  { int waves = (2 * NPIX / 32) * 10;
    h1_gemm<<<(waves * 32 + 255) / 256, 256, 0, stream>>>(x1, x2, cw1, s1, t1, h1); }
  // h2 = relu(bn2(h1))
  bn2_relu<<<(2 * 32 * NPIX + 255) / 256, 256, 0, stream>>>(h1, g2, bt2, m2, v2p, h2);
  // wgt = cw2 @ h2 + cb2   (C=32, O=1568)
  { int waves = (1568 / 32) * (2 * NPIX / 64);
    conv1x1_wmma<4><<<(waves * 32 + 255) / 256, 256, 0, stream>>>(h2, cw2, cb2, wgt, 32, 1568); }
  // final 49-tap aggregation
  aggregate<<<(2 * 256 * NPIX + 255) / 256, 256, 0, stream>>>(x3, wgt, out);
}